// DecGreenNet_product_tucker2_90967407329578
// MI455X (gfx1250) — compile-verified
//
#include <hip/hip_runtime.h>
#include <math.h>

#define PI_F 3.14159265358979323846f

typedef __attribute__((ext_vector_type(2))) float v2f;
typedef __attribute__((ext_vector_type(8))) float v8f;

__device__ __forceinline__ float fast_tanh(float x) {
#if __has_builtin(__builtin_amdgcn_tanhf)
  return __builtin_amdgcn_tanhf(x);        // gfx1250 v_tanh_f32
#elif __has_builtin(__builtin_amdgcn_tanh_f32)
  return __builtin_amdgcn_tanh_f32(x);
#else
  return tanhf(x);
#endif
}

// ws layout (floats): [0,64) s0 | [64,128) s1 | [128,256) hc | [256,320) core3
//                     [320,576) v | [576] c     (total 577 floats)

__global__ void k0_zero(float* ws) {
  int t = blockIdx.x * blockDim.x + threadIdx.x;
  if (t < 577) ws[t] = 0.0f;
}

// blocks 0..15: axis0 quad chunks -> s0 ; blocks 16..31: axis1 -> s1 ; block 32: hc
__global__ void k1_setup(const float* __restrict__ eq,
                         const float* __restrict__ qx0, const float* __restrict__ qx1,
                         const float* __restrict__ cinit,
                         const float* __restrict__ Wq01, const float* __restrict__ bq01,
                         const float* __restrict__ Wq02, const float* __restrict__ bq02,
                         const float* __restrict__ Wq11, const float* __restrict__ bq11,
                         const float* __restrict__ Wq12, const float* __restrict__ bq12,
                         const float* __restrict__ Wc1,  const float* __restrict__ bc1,
                         float* ws) {
  __shared__ float th[64 * 128];   // tanh(layer1) for 64 quad points
  __shared__ float red[256];
  const int t = threadIdx.x;
  const int blk = blockIdx.x;

  if (blk == 32) {                 // hc = tanh(core_init @ Wc1 + bc1)
    if (t < 128) {
      float acc = bc1[t];
      for (int i = 0; i < 64; ++i) acc += cinit[i] * Wc1[i * 128 + t];
      ws[128 + t] = fast_tanh(acc);
    }
    return;
  }

  const int ax    = blk >> 4;      // 0 or 1
  const int chunk = blk & 15;      // 0..15 (64 quad points each)
  const float* q  = ax ? qx1 : qx0;
  const float* W1 = ax ? Wq11 : Wq01;
  const float* b1 = ax ? bq11 : bq01;
  const float* W2 = ax ? Wq12 : Wq02;
  const float* b2 = ax ? bq12 : bq02;
  float* sdst = ws + (ax ? 64 : 0);
  const int c0 = chunk * 64;

  // layer 1 + tanh for this chunk's 64 quad points (64 x 128)
  for (int e = 0; e < 32; ++e) {
    int flat = t + e * 256;
    int ci = flat >> 7, j = flat & 127;
    float x = q[c0 + ci];
    th[flat] = fast_tanh(x * W1[j] + b1[j]);
  }
  __syncthreads();

  // s[r] += sum_c y(c) * ( th[c,:] . W2[:,r] + b2[r] )
  const int r = t & 63, g = t >> 6;
  const float eqv = eq[0];
  float psum = 0.0f;
  for (int ci = g * 16; ci < g * 16 + 16; ++ci) {
    float x = q[c0 + ci];
    float y = __sinf(PI_F * eqv * x);
    float dot = b2[r];
    const float* thc = th + ci * 128;
    for (int j = 0; j < 128; ++j) dot += thc[j] * W2[j * 64 + r];
    psum += y * dot;
  }
  red[t] = psum;
  __syncthreads();
  if (g == 0)
    atomicAdd(&sdst[r], red[r] + red[r + 64] + red[r + 128] + red[r + 192]);
}

// core3[a] = sum_{j,k} hc[j]*Wc2[j, a*4096+k]*w[k] + sum_k bc2[a*4096+k]*w[k]
// w[k] = s0[k>>6]*s1[k&63]
// blocks 0..511: Wc2 part (a = blk>>3, j-slice = blk&7); blocks 512..575: bias part
// k = kk*1024 + 4t + i : k>>6 = kk*16 + (t>>4) (same for i=0..3); k&63 = (4t+i)&63
__global__ void k2_core(const float* __restrict__ Wc2, const float* __restrict__ bc2,
                        float* ws) {
  __shared__ float s0s[64], s1s[64], hcs[16], red[256];
  const int t = threadIdx.x;
  const int blk = blockIdx.x;
  const bool isBias = (blk >= 512);
  const int a   = isBias ? (blk - 512) : (blk >> 3);
  const int sub = isBias ? 0 : (blk & 7);

  if (t < 64) { s0s[t] = ws[t]; s1s[t] = ws[64 + t]; }
  if (t < 16) hcs[t] = ws[128 + sub * 16 + t];
  __syncthreads();

  const int s0base = t >> 4;
  float s1v[4];
#pragma unroll
  for (int i = 0; i < 4; ++i) s1v[i] = s1s[(4 * t + i) & 63];

  float acc = 0.0f;
  if (!isBias) {
    const size_t colbase = (size_t)a * 4096 + 4 * t;
    for (int jj = 0; jj < 16; ++jj) {
      const float* base = Wc2 + (size_t)(sub * 16 + jj) * 262144 + colbase;
      float pj = 0.0f;
#pragma unroll
      for (int kk = 0; kk < 4; ++kk) {
        const float4 w = *(const float4*)(base + kk * 1024);
        pj += s0s[kk * 16 + s0base] *
              (w.x * s1v[0] + w.y * s1v[1] + w.z * s1v[2] + w.w * s1v[3]);
      }
      acc += hcs[jj] * pj;
    }
  } else {
    const float* base = bc2 + (size_t)a * 4096 + 4 * t;
#pragma unroll
    for (int kk = 0; kk < 4; ++kk) {
      const float4 w = *(const float4*)(base + kk * 1024);
      acc += s0s[kk * 16 + s0base] *
             (w.x * s1v[0] + w.y * s1v[1] + w.z * s1v[2] + w.w * s1v[3]);
    }
  }
  red[t] = acc;
  __syncthreads();
  for (int s = 128; s > 0; s >>= 1) {
    if (t < s) red[t] += red[t + s];
    __syncthreads();
  }
  if (t == 0) atomicAdd(&ws[256 + a], red[0]);
}

// v = Wx2 @ core3 (256), c = bx2 . core3
__global__ void k3_v(const float* __restrict__ Wx2, const float* __restrict__ bx2,
                     float* ws) {
  const int t = threadIdx.x;
  const float* core3 = ws + 256;
  float acc = 0.0f;
  for (int a = 0; a < 64; ++a) acc += Wx2[t * 64 + a] * core3[a];
  ws[320 + t] = acc;
  if (t == 0) {
    float cc = 0.0f;
    for (int a = 0; a < 64; ++a) cc += bx2[a] * core3[a];
    ws[576] = cc;
  }
}

// out[n] = tanh(input[n] @ Wx1 + bx1) . v + c
// WMMA f32 16x16x4: K=0,1 carry (x0,x1)x(W0,W1); K=2 carries the bias
// (A row K=2 = 1.0, B row K=2 = bx1[col]); K=3 zero. C = inline 0.
__global__ void __launch_bounds__(256) k4_main(const float* __restrict__ input,
                                               const float* __restrict__ Wx1,
                                               const float* __restrict__ bx1,
                                               const float* __restrict__ ws,
                                               float* __restrict__ out) {
  // packed per-(tile,lane) operands: {B_vgpr0, B_vgpr1, v[col], 0}
  __shared__ float4 sP[512];
  const int t = threadIdx.x;
  for (int idx = t; idx < 512; idx += 256) {
    const int tl = idx >> 5;            // tile 0..15
    const int ln = idx & 31;
    const int col = tl * 16 + (ln & 15);
    float4 p;
    if (ln < 16) { p.x = Wx1[col]; p.y = Wx1[256 + col]; }  // K=0, K=1 rows
    else         { p.x = bx1[col]; p.y = 0.0f; }            // K=2 (bias), K=3
    p.z = ws[320 + col];                                    // v[col]
    p.w = 0.0f;
    sP[idx] = p;
  }
  __syncthreads();

  const float cval = ws[576];
  const int lane = t & 31;
  const int wave = t >> 5;

  for (int g = 0; g < 4; ++g) {
    const int pbase = blockIdx.x * 512 + wave * 64 + g * 16;   // 16 pts per group

    float a0, a1;
    if (lane < 16) {
      const float* ip = input + 2 * (size_t)(pbase + lane);
      a0 = ip[0]; a1 = ip[1];            // K=0, K=1
    } else {
      a0 = 1.0f; a1 = 0.0f;              // K=2 = 1 (bias row), K=3 = 0
    }
    v2f A; A[0] = a0; A[1] = a1;

    float acc[8];
#pragma unroll
    for (int m = 0; m < 8; ++m) acc[m] = 0.0f;

#pragma unroll
    for (int tile = 0; tile < 16; ++tile) {
      const float4 p = sP[tile * 32 + lane];
      v2f B; B[0] = p.x; B[1] = p.y;
      v8f C = {};
      v8f D = __builtin_amdgcn_wmma_f32_16x16x4_f32(
          false, A, false, B, (short)0, C, false, false);
#pragma unroll
      for (int m = 0; m < 8; ++m) acc[m] += fast_tanh(D[m]) * p.z;
    }

    // reduce 16 columns within each 16-lane half (rows M=0..7 / M=8..15)
#pragma unroll
    for (int m = 0; m < 8; ++m) {
      float v = acc[m];
      v += __shfl_xor(v, 1, 32);
      v += __shfl_xor(v, 2, 32);
      v += __shfl_xor(v, 4, 32);
      v += __shfl_xor(v, 8, 32);
      acc[m] = v;
    }
    if ((lane & 15) == 0) {
      float* o = out + pbase + (lane >> 4) * 8;
#pragma unroll
      for (int m = 0; m < 8; ++m) o[m] = acc[m] + cval;
    }
  }
}

extern "C" void kernel_launch(void* const* d_in, const int* in_sizes, int n_in,
                              void* d_out, int out_size, void* d_ws, size_t ws_size,
                              hipStream_t stream) {
  const float* input = (const float*)d_in[0];
  const float* eq    = (const float*)d_in[1];
  const float* qx0   = (const float*)d_in[2];
  const float* qx1   = (const float*)d_in[3];
  const float* cinit = (const float*)d_in[4];
  const float* Wx1   = (const float*)d_in[5];
  const float* bx1   = (const float*)d_in[6];
  const float* Wx2   = (const float*)d_in[7];
  const float* bx2   = (const float*)d_in[8];
  const float* Wq01  = (const float*)d_in[9];
  const float* bq01  = (const float*)d_in[10];
  const float* Wq02  = (const float*)d_in[11];
  const float* bq02  = (const float*)d_in[12];
  const float* Wq11  = (const float*)d_in[13];
  const float* bq11  = (const float*)d_in[14];
  const float* Wq12  = (const float*)d_in[15];
  const float* bq12  = (const float*)d_in[16];
  const float* Wc1   = (const float*)d_in[17];
  const float* bc1   = (const float*)d_in[18];
  const float* Wc2   = (const float*)d_in[19];
  const float* bc2   = (const float*)d_in[20];
  float* ws  = (float*)d_ws;
  float* out = (float*)d_out;
  const int N = in_sizes[0] / 2;

  hipLaunchKernelGGL(k0_zero, dim3(3), dim3(256), 0, stream, ws);
  hipLaunchKernelGGL(k1_setup, dim3(33), dim3(256), 0, stream,
                     eq, qx0, qx1, cinit, Wq01, bq01, Wq02, bq02,
                     Wq11, bq11, Wq12, bq12, Wc1, bc1, ws);
  hipLaunchKernelGGL(k2_core, dim3(576), dim3(256), 0, stream, Wc2, bc2, ws);
  hipLaunchKernelGGL(k3_v, dim3(1), dim3(256), 0, stream, Wx2, bx2, ws);
  hipLaunchKernelGGL(k4_main, dim3(N / 512), dim3(256), 0, stream,
                     input, Wx1, bx1, ws, out);
}